// IWsoftCrossEntropy_86517821215224
// MI455X (gfx1250) — compile-verified
//
#include <hip/hip_runtime.h>
#include <math.h>

// Problem constants (fixed by the reference's setup_inputs)
#define NCLS        19
#define HWPIX       (1024 * 2048)          // pixels per image, N==1
#define TILE        256                    // pixels staged per tile
#define BLOCK       128                    // threads per block (4 waves)
#define ROWS        (2 * NCLS)             // 19 input rows + 19 target rows
#define CPR         (TILE / 4)             // 16B chunks per row = 64
#define CHUNKS      (ROWS * CPR)           // 2432 chunks per tile
#define PER_THREAD  (CHUNKS / BLOCK)       // 19 async loads per thread/tile
#define NBLOCKS     1024
#define PIX_PER_BLK (HWPIX / NBLOCKS)      // 2048
#define TILES_PER_BLK (PIX_PER_BLK / TILE) // 8

__global__ __launch_bounds__(BLOCK)
void iw_ce_main(const float* __restrict__ in,
                const float* __restrict__ tg,
                float* __restrict__ gBsum,
                unsigned* __restrict__ gHist)
{
    // 2-stage double buffer: [stage][row][pixel], rows 0..18 = inputs ch,
    // rows 19..37 = target ch.  2*38*256*4 = 77824 B static LDS.
    __shared__ float    smem[2][ROWS][TILE];
    __shared__ float    bsum_s[NCLS];
    __shared__ unsigned hist_s[NCLS];

    const int tid       = threadIdx.x;
    const int block_px0 = blockIdx.x * PIX_PER_BLK;

    if (tid < NCLS) { bsum_s[tid] = 0.0f; hist_s[tid] = 0u; }

    // Issue one tile's worth of async global->LDS DMA. Exactly PER_THREAD
    // (=19) instructions per lane, uniform across the wave, so the wave's
    // ASYNCcnt advances by 19 per tile.
    auto issue_tile = [&](int t) {
        const int stage = t & 1;
        const int px0   = block_px0 + t * TILE;
        unsigned  lbase = (unsigned)(size_t)(&smem[stage][0][0]);
#pragma unroll
        for (int k = 0; k < PER_THREAD; ++k) {
            int i   = tid + k * BLOCK;
            int row = i >> 6;               // / CPR
            int ch  = (i & (CPR - 1)) << 2; // chunk -> first pixel in chunk
            const float* src = (row < NCLS)
                ? (in + (size_t)row * HWPIX + px0 + ch)
                : (tg + (size_t)(row - NCLS) * HWPIX + px0 + ch);
            unsigned ldst = lbase + (unsigned)((row * TILE + ch) * 4);
            asm volatile("global_load_async_to_lds_b128 %0, %1, off"
                         :: "v"(ldst), "v"(src)
                         : "memory");
        }
    };

    issue_tile(0);

    for (int t = 0; t < TILES_PER_BLK; ++t) {
        if (t + 1 < TILES_PER_BLK) {
            issue_tile(t + 1);
            // Older tile's 19 loads retired when <=19 remain outstanding
            // (async loads complete in order within a wave).
            asm volatile("s_wait_asynccnt 19" ::: "memory");
        } else {
            asm volatile("s_wait_asynccnt 0" ::: "memory");
        }
        __syncthreads();   // all waves' DMA for tile t visible in LDS

        const int stage = t & 1;
#pragma unroll
        for (int k = 0; k < TILE / BLOCK; ++k) {
            const int    px = tid + k * BLOCK;
            const float* bi = &smem[stage][0][px];     // stride TILE per ch
            const float* bt = &smem[stage][NCLS][px];

            // Online softmax + argmax + masked weighted dot, one pass over c.
            float x0  = bi[0];
            float m   = x0;
            int   arg = 0;
            float s   = 1.0f;                 // exp(x0 - m)
            float t0  = bt[0];
            float mt0 = (t0 != -1.0f) ? t0 : 0.0f;
            float Dv  = mt0 * x0;
            float Tv  = mt0;
#pragma unroll
            for (int c = 1; c < NCLS; ++c) {
                float x  = bi[c * TILE];
                float tv = bt[c * TILE];
                float mn = fmaxf(m, x);
                s   = s * __expf(m - mn) + __expf(x - mn);
                arg = (x > m) ? c : arg;      // strict '>' keeps first max
                m   = mn;
                float mtc = (tv != -1.0f) ? tv : 0.0f;
                Dv = fmaf(mtc, x, Dv);
                Tv += mtc;
            }
            float lse = m + __logf(s);
            float A   = Dv - lse * Tv;        // sum_c mask*t*(x - lse)

            atomicAdd(&hist_s[arg], 1u);
            atomicAdd(&bsum_s[arg], A);
        }
        __syncthreads();   // buffer (t&1) free for tile t+2's DMA
    }

    if (tid < NCLS) {
        atomicAdd(&gHist[tid], hist_s[tid]);
        atomicAdd(&gBsum[tid], bsum_s[tid]);
    }
}

__global__ void iw_ce_zero(float* gBsum, unsigned* gHist)
{
    int i = threadIdx.x;
    if (i < NCLS) { gBsum[i] = 0.0f; gHist[i] = 0u; }
}

__global__ void iw_ce_final(const float* __restrict__ gBsum,
                            const unsigned* __restrict__ gHist,
                            float* __restrict__ out)
{
    if (threadIdx.x == 0) {
        float total = 0.0f;
        for (int c = 0; c < NCLS; ++c) total += (float)gHist[c];
        float acc = 0.0f;
        for (int c = 0; c < NCLS; ++c) {
            float h = (float)gHist[c];
            float w = 1.0f / fmaxf(powf(h, 0.2f) * powf(total, 0.8f), 1.0f);
            acc += gBsum[c] * w;
        }
        out[0] = -acc / (float)NCLS;   // sum(weighted)/(N*C), N==1
    }
}

extern "C" void kernel_launch(void* const* d_in, const int* in_sizes, int n_in,
                              void* d_out, int out_size, void* d_ws, size_t ws_size,
                              hipStream_t stream)
{
    const float* in  = (const float*)d_in[0];   // inputs (1,19,1024,2048) f32
    const float* tg  = (const float*)d_in[1];   // target (1,19,1024,2048) f32
    float*    gBsum  = (float*)d_ws;                         // 19 floats
    unsigned* gHist  = (unsigned*)((char*)d_ws + NCLS * 4);  // 19 uints
    float*    out    = (float*)d_out;

    iw_ce_zero <<<1, 32, 0, stream>>>(gBsum, gHist);
    iw_ce_main <<<NBLOCKS, BLOCK, 0, stream>>>(in, tg, gBsum, gHist);
    iw_ce_final<<<1, 1, 0, stream>>>(gBsum, gHist, out);
}